// TensorEmbedding_12008728560153
// MI455X (gfx1250) — compile-verified
//
#include <hip/hip_runtime.h>

#define NN 10000   // nodes
#define HH 128     // hidden
#define EE 64000   // edges
#define RR 64      // num_rbf

typedef float v2f __attribute__((ext_vector_type(2)));
typedef float v8f __attribute__((ext_vector_type(8)));

// ---------------- WMMA helpers (V_WMMA_F32_16X16X4_F32, wave32) ----------------

static __device__ __forceinline__ v8f wmma4(v2f a, v2f b, v8f c) {
  // 8 args: (neg_a, A, neg_b, B, c_mod, C, reuse_a, reuse_b)
  return __builtin_amdgcn_wmma_f32_16x16x4_f32(false, a, false, b, (short)0, c,
                                               false, false);
}

// A fragment: 16x4 tile of row-major matrix p (leading dim ld), rows = M.
// VGPR0: lanes0-15 K=k0, lanes16-31 K=k0+2 ; VGPR1: K=k0+1 / K=k0+3.
static __device__ __forceinline__ v2f loadA_ld(const float* p, int ld, int k0, int lane) {
  int row = lane & 15;
  int kk  = k0 + ((lane >> 4) << 1);
  v2f a;
  a.x = p[row * ld + kk];
  a.y = p[row * ld + kk + 1];
  return a;
}

// A fragment with explicit row/col strides (for [16,H,comp] interleaved LDS tiles)
static __device__ __forceinline__ v2f loadA_str(const float* p, int rs, int cs, int k0, int lane) {
  int row = lane & 15;
  int kk  = k0 + ((lane >> 4) << 1);
  v2f a;
  a.x = p[row * rs + kk * cs];
  a.y = p[row * rs + kk * cs + cs];
  return a;
}

// B fragment (4x16, K x N) taken from weight matrix W[g, k] (out = in @ W^T):
// B[k][n] = W[g0+n][k].
static __device__ __forceinline__ v2f loadB_gk(const float* W, int ld, int g0, int k0, int lane) {
  int g  = g0 + (lane & 15);
  int kk = k0 + ((lane >> 4) << 1);
  v2f b;
  b.x = W[g * ld + kk];
  b.y = W[g * ld + kk + 1];
  return b;
}

static __device__ __forceinline__ float silu(float x) {
  return x / (1.0f + __expf(-x));
}

// ---------------- async copy: global -> LDS, 512B per wave-instruction ----------------
// Each of the 32 lanes moves 16B; tracked by ASYNCcnt.

static __device__ __forceinline__ void async_chunk512(unsigned lds_base,
                                                      const float* gbase,
                                                      int chunk, int lane) {
  unsigned long long g =
      (unsigned long long)(size_t)gbase + (unsigned)(chunk * 512 + lane * 16);
  unsigned l = lds_base + (unsigned)(chunk * 512 + lane * 16);
  asm volatile("global_load_async_to_lds_b128 %0, %1, off"
               :: "v"(l), "v"(g) : "memory");
}

static __device__ __forceinline__ void async_wait0() {
  asm volatile("s_wait_asynccnt 0x0" ::: "memory");
}

// ---------------- kernel 0: zero the atomic accumulators ----------------

__global__ void zero_kernel(float4* p, int n4) {
  int i = blockIdx.x * blockDim.x + threadIdx.x;
  if (i < n4) p[i] = make_float4(0.f, 0.f, 0.f, 0.f);
}

// ---------------- kernel 1: node embedding Z[n,h] ----------------

__global__ void embed_kernel(const int* __restrict__ x,
                             const int* __restrict__ ascal,
                             const float* __restrict__ emb_z,
                             const float* __restrict__ ae_emb,
                             float* __restrict__ Z) {
  int idx = blockIdx.x * blockDim.x + threadIdx.x;
  if (idx >= NN * HH) return;
  int n = idx >> 7;
  int h = idx & 127;
  const int offs[9] = {0, 10, 30, 40, 50, 70, 90, 100, 110};
  float acc = emb_z[x[n] * HH + h];
#pragma unroll
  for (int j = 0; j < 9; ++j)
    acc += ae_emb[(ascal[n * 9 + j] + offs[j]) * HH + h];
  Z[idx] = acc;
}

// ---------------- kernel 2: edge GEMMs (WMMA) + scatter of 10 moments ----------------
// block = 128 threads (4 waves), 16 edges per block, 8 h-tiles of 16.

__global__ void edge_kernel(const float* __restrict__ Z,
                            const int* __restrict__ src,
                            const int* __restrict__ dst,
                            const float* __restrict__ rbf,
                            const float* __restrict__ vecn,
                            const float* __restrict__ emb2_w,  // [128,256]
                            const float* __restrict__ emb2_b,
                            const float* __restrict__ dp1_w,   // [128,64]
                            const float* __restrict__ dp1_b,
                            const float* __restrict__ dp2_w,
                            const float* __restrict__ dp2_b,
                            const float* __restrict__ dp3_w,
                            const float* __restrict__ dp3_b,
                            float* __restrict__ nodeI,         // [N,128]
                            float* __restrict__ nodeW,         // [N,128,3]
                            float* __restrict__ nodeS) {       // [N,128,6]
  __shared__ float4 sZsrc4[16 * 32];  // 16 x 128 floats
  __shared__ float4 sZdst4[16 * 32];
  __shared__ float4 sRbf4[16 * 16];   // 16 x 64 floats (contiguous rows in global)
  __shared__ float  sV[16 * 3];
  __shared__ float  sSym[16 * 6];
  __shared__ int    sDst[16];
  float* sZsrc = (float*)sZsrc4;
  float* sZdst = (float*)sZdst4;
  float* sRbf  = (float*)sRbf4;

  const int e0   = blockIdx.x * 16;
  const int tid  = threadIdx.x;
  const int wave = tid >> 5;
  const int lane = tid & 31;

  if (tid < 16) {
    int e = e0 + tid;
    sDst[tid] = dst[e];
    float v0 = vecn[e * 3 + 0], v1 = vecn[e * 3 + 1], v2 = vecn[e * 3 + 2];
    sV[tid * 3 + 0] = v0; sV[tid * 3 + 1] = v1; sV[tid * 3 + 2] = v2;
    float tr3 = (v0 * v0 + v1 * v1 + v2 * v2) * (1.0f / 3.0f);
    sSym[tid * 6 + 0] = v0 * v0 - tr3;  // 00
    sSym[tid * 6 + 1] = v0 * v1;        // 01
    sSym[tid * 6 + 2] = v0 * v2;        // 02
    sSym[tid * 6 + 3] = v1 * v1 - tr3;  // 11
    sSym[tid * 6 + 4] = v1 * v2;        // 12
    sSym[tid * 6 + 5] = v2 * v2 - tr3;  // 22
  }

  // async-stage the gathered Z rows: each wave moves 4 rows of Zsrc + 4 of Zdst
  {
    const unsigned zsrc_off = (unsigned)(size_t)sZsrc4;
    const unsigned zdst_off = (unsigned)(size_t)sZdst4;
#pragma unroll
    for (int j = 0; j < 4; ++j) {
      int r  = wave * 4 + j;
      int es = src[e0 + r];
      int ed = dst[e0 + r];
      async_chunk512(zsrc_off + (unsigned)(r * 512), Z + (size_t)es * HH, 0, lane);
      async_chunk512(zdst_off + (unsigned)(r * 512), Z + (size_t)ed * HH, 0, lane);
    }
    // rbf tile is a contiguous 4KB region (rows e0..e0+15): 8 chunks of 512B
    const unsigned rbf_off = (unsigned)(size_t)sRbf4;
    const float*   rbf_g   = rbf + (size_t)e0 * RR;
    async_chunk512(rbf_off, rbf_g, wave, lane);
    async_chunk512(rbf_off, rbf_g, wave + 4, lane);
  }
  if (tid == 0) {  // warm the weight lines (global_prefetch_b8)
    __builtin_prefetch(emb2_w, 0, 0);
    __builtin_prefetch(dp1_w, 0, 0);
    __builtin_prefetch(dp2_w, 0, 0);
    __builtin_prefetch(dp3_w, 0, 0);
  }
  async_wait0();
  __syncthreads();

  for (int ht = wave; ht < 8; ht += 4) {
    const int g0 = ht * 16;
    v8f zij = {};
#pragma unroll 4
    for (int k0 = 0; k0 < HH; k0 += 4) {
      zij = wmma4(loadA_ld(sZsrc, HH, k0, lane),
                  loadB_gk(emb2_w, 2 * HH, g0, k0, lane), zij);
      zij = wmma4(loadA_ld(sZdst, HH, k0, lane),
                  loadB_gk(emb2_w, 2 * HH, g0, HH + k0, lane), zij);
    }
    v8f p1 = {}, p2 = {}, p3 = {};
#pragma unroll 4
    for (int k0 = 0; k0 < RR; k0 += 4) {
      v2f a = loadA_ld(sRbf, RR, k0, lane);
      p1 = wmma4(a, loadB_gk(dp1_w, RR, g0, k0, lane), p1);
      p2 = wmma4(a, loadB_gk(dp2_w, RR, g0, k0, lane), p2);
      p3 = wmma4(a, loadB_gk(dp3_w, RR, g0, k0, lane), p3);
    }
    const int h  = g0 + (lane & 15);
    const int mo = (lane >> 4) << 3;
    const float zb = emb2_b[h];
    const float b1 = dp1_b[h], b2 = dp2_b[h], b3 = dp3_b[h];
#pragma unroll
    for (int i = 0; i < 8; ++i) {
      int m = i + mo;
      float zv = zij[i] + zb;
      float c1 = (p1[i] + b1) * zv;
      float c2 = (p2[i] + b2) * zv;
      float c3 = (p3[i] + b3) * zv;
      int d = sDst[m];
      atomicAdd(&nodeI[(size_t)d * HH + h], c1);
      float* wp = &nodeW[((size_t)d * HH + h) * 3];
      atomicAdd(wp + 0, c2 * sV[m * 3 + 0]);
      atomicAdd(wp + 1, c2 * sV[m * 3 + 1]);
      atomicAdd(wp + 2, c2 * sV[m * 3 + 2]);
      float* sp = &nodeS[((size_t)d * HH + h) * 6];
      atomicAdd(sp + 0, c3 * sSym[m * 6 + 0]);
      atomicAdd(sp + 1, c3 * sSym[m * 6 + 1]);
      atomicAdd(sp + 2, c3 * sSym[m * 6 + 2]);
      atomicAdd(sp + 3, c3 * sSym[m * 6 + 3]);
      atomicAdd(sp + 4, c3 * sSym[m * 6 + 4]);
      atomicAdd(sp + 5, c3 * sSym[m * 6 + 5]);
    }
  }
}

// ---------------- kernel 3: Frobenius norm^2 + LayerNorm ----------------
// ||T||^2 = 3 i^2 + 2|w|^2 + s00^2+s11^2+s22^2 + 2(s01^2+s02^2+s12^2)

__global__ void norm_kernel(const float* __restrict__ nodeI,
                            const float* __restrict__ nodeW,
                            const float* __restrict__ nodeS,
                            const float* __restrict__ ln_g,
                            const float* __restrict__ ln_b,
                            float* __restrict__ lnout) {
  __shared__ float red[HH];
  int n = blockIdx.x, h = threadIdx.x;
  float iv = nodeI[(size_t)n * HH + h];
  const float* w = &nodeW[((size_t)n * HH + h) * 3];
  const float* s = &nodeS[((size_t)n * HH + h) * 6];
  float f = 3.f * iv * iv
          + 2.f * (w[0] * w[0] + w[1] * w[1] + w[2] * w[2])
          + s[0] * s[0] + s[3] * s[3] + s[5] * s[5]
          + 2.f * (s[1] * s[1] + s[2] * s[2] + s[4] * s[4]);
  red[h] = f;
  __syncthreads();
  for (int st = 64; st > 0; st >>= 1) {
    if (h < st) red[h] += red[h + st];
    __syncthreads();
  }
  float mu = red[0] * (1.f / HH);
  __syncthreads();
  float d = f - mu;
  red[h] = d * d;
  __syncthreads();
  for (int st = 64; st > 0; st >>= 1) {
    if (h < st) red[h] += red[h + st];
    __syncthreads();
  }
  float var = red[0] * (1.f / HH);
  lnout[(size_t)n * HH + h] = d * rsqrtf(var + 1e-5f) * ln_g[h] + ln_b[h];
}

// ---------------- kernel 4: fused MLP (ls1 silu -> ls2 silu) via WMMA ----------------
// block = 256 threads (8 waves), 16 nodes per block.

__global__ void mlp_kernel(const float* __restrict__ lnout,
                           const float* __restrict__ ls1_w,  // [256,128]
                           const float* __restrict__ ls1_b,
                           const float* __restrict__ ls2_w,  // [384,256]
                           const float* __restrict__ ls2_b,
                           float* __restrict__ norm3) {      // [N,384]
  __shared__ float4 sIn4[16 * 32];    // 16 x 128
  __shared__ float4 sY14[16 * 64];    // 16 x 256
  float* sIn = (float*)sIn4;
  float* sY1 = (float*)sY14;
  const int n0   = blockIdx.x * 16;
  const int wave = threadIdx.x >> 5;
  const int lane = threadIdx.x & 31;

  // async-stage the contiguous 8KB input tile: 16 chunks over 8 waves
  {
    const unsigned in_off = (unsigned)(size_t)sIn4;
    const float*   g      = lnout + (size_t)n0 * HH;
    async_chunk512(in_off, g, wave, lane);
    async_chunk512(in_off, g, wave + 8, lane);
  }
  async_wait0();
  __syncthreads();

  // y1 = silu(lnout @ ls1_w^T + b) : 16 tiles of 16 over 256 outputs
  for (int t = wave; t < 16; t += 8) {
    const int g0 = t * 16;
    v8f acc = {};
#pragma unroll 4
    for (int k0 = 0; k0 < HH; k0 += 4)
      acc = wmma4(loadA_ld(sIn, HH, k0, lane),
                  loadB_gk(ls1_w, HH, g0, k0, lane), acc);
    const int g  = g0 + (lane & 15);
    const int mo = (lane >> 4) << 3;
    const float bv = ls1_b[g];
#pragma unroll
    for (int i = 0; i < 8; ++i)
      sY1[(i + mo) * 256 + g] = silu(acc[i] + bv);
  }
  __syncthreads();
  // y2 = silu(y1 @ ls2_w^T + b) : 24 tiles of 16 over 384 outputs
  for (int t = wave; t < 24; t += 8) {
    const int g0 = t * 16;
    v8f acc = {};
#pragma unroll 4
    for (int k0 = 0; k0 < 256; k0 += 4)
      acc = wmma4(loadA_ld(sY1, 256, k0, lane),
                  loadB_gk(ls2_w, 256, g0, k0, lane), acc);
    const int g  = g0 + (lane & 15);
    const int mo = (lane >> 4) << 3;
    const float bv = ls2_b[g];
#pragma unroll
    for (int i = 0; i < 8; ++i)
      norm3[(size_t)(n0 + i + mo) * 384 + g] = silu(acc[i] + bv);
  }
}

// ---------------- kernel 5: component GEMMs (lt0/lt1/lt2) + tensor assembly ----------------
// block = 256 threads (8 waves), 16 nodes; each wave owns one g-tile, 10 accumulators.
// A-operands staged to LDS with async copies (80KB: fine vs 320KB/WGP).

__global__ void out_kernel(const float* __restrict__ nodeI,
                           const float* __restrict__ nodeW,
                           const float* __restrict__ nodeS,
                           const float* __restrict__ lt0,   // [128,128]
                           const float* __restrict__ lt1,
                           const float* __restrict__ lt2,
                           const float* __restrict__ norm3, // [N,384]
                           float* __restrict__ out) {       // [N,128,3,3]
  __shared__ float4 sI4[16 * 32];     // 16 x 128        ( 8KB)
  __shared__ float4 sW4[16 * 96];     // 16 x 128 x 3    (24KB)
  __shared__ float4 sS4[16 * 192];    // 16 x 128 x 6    (48KB)
  float* sI = (float*)sI4;
  float* sW = (float*)sW4;
  float* sS = (float*)sS4;

  const int n0   = blockIdx.x * 16;
  const int wave = threadIdx.x >> 5;
  const int lane = threadIdx.x & 31;
  const int g0   = wave * 16;

  // contiguous async staging: 16 + 48 + 96 = 160 chunks of 512B over 8 waves
  {
    const unsigned iof = (unsigned)(size_t)sI4;
    const unsigned wof = (unsigned)(size_t)sW4;
    const unsigned sof = (unsigned)(size_t)sS4;
    const float* gI = nodeI + (size_t)n0 * HH;
    const float* gW = nodeW + (size_t)n0 * HH * 3;
    const float* gS = nodeS + (size_t)n0 * HH * 6;
    for (int c = wave; c < 16; c += 8)  async_chunk512(iof, gI, c, lane);
    for (int c = wave; c < 48; c += 8)  async_chunk512(wof, gW, c, lane);
    for (int c = wave; c < 96; c += 8)  async_chunk512(sof, gS, c, lane);
  }
  async_wait0();
  __syncthreads();

  v8f aI = {};
  v8f aW0 = {}, aW1 = {}, aW2 = {};
  v8f aS0 = {}, aS1 = {}, aS2 = {}, aS3 = {}, aS4 = {}, aS5 = {};

#pragma unroll 2
  for (int k0 = 0; k0 < HH; k0 += 4) {
    v2f bI = loadB_gk(lt0, HH, g0, k0, lane);
    aI = wmma4(loadA_ld(sI, HH, k0, lane), bI, aI);
    v2f bW = loadB_gk(lt1, HH, g0, k0, lane);
    aW0 = wmma4(loadA_str(sW + 0, 3 * HH, 3, k0, lane), bW, aW0);
    aW1 = wmma4(loadA_str(sW + 1, 3 * HH, 3, k0, lane), bW, aW1);
    aW2 = wmma4(loadA_str(sW + 2, 3 * HH, 3, k0, lane), bW, aW2);
    v2f bS = loadB_gk(lt2, HH, g0, k0, lane);
    aS0 = wmma4(loadA_str(sS + 0, 6 * HH, 6, k0, lane), bS, aS0);
    aS1 = wmma4(loadA_str(sS + 1, 6 * HH, 6, k0, lane), bS, aS1);
    aS2 = wmma4(loadA_str(sS + 2, 6 * HH, 6, k0, lane), bS, aS2);
    aS3 = wmma4(loadA_str(sS + 3, 6 * HH, 6, k0, lane), bS, aS3);
    aS4 = wmma4(loadA_str(sS + 4, 6 * HH, 6, k0, lane), bS, aS4);
    aS5 = wmma4(loadA_str(sS + 5, 6 * HH, 6, k0, lane), bS, aS5);
  }

  const int g  = g0 + (lane & 15);
  const int mo = (lane >> 4) << 3;
#pragma unroll
  for (int i = 0; i < 8; ++i) {
    int n = n0 + i + mo;
    const float* nm = norm3 + (size_t)n * 384 + g * 3;
    float na = nm[0], nb = nm[1], nc = nm[2];
    float it  = aI[i] * na;
    float w0  = aW0[i] * nb, w1 = aW1[i] * nb, w2 = aW2[i] * nb;
    float s00 = aS0[i] * nc, s01 = aS1[i] * nc, s02 = aS2[i] * nc;
    float s11 = aS3[i] * nc, s12 = aS4[i] * nc, s22 = aS5[i] * nc;
    float* o = out + ((size_t)n * HH + g) * 9;
    o[0] = it + s00;  o[1] = -w2 + s01; o[2] =  w1 + s02;
    o[3] =  w2 + s01; o[4] = it + s11;  o[5] = -w0 + s12;
    o[6] = -w1 + s02; o[7] =  w0 + s12; o[8] = it + s22;
  }
}

// ---------------- launcher ----------------

extern "C" void kernel_launch(void* const* d_in, const int* in_sizes, int n_in,
                              void* d_out, int out_size, void* d_ws, size_t ws_size,
                              hipStream_t stream) {
  const int*   x       = (const int*)d_in[0];
  const int*   ascal   = (const int*)d_in[1];
  const int*   src     = (const int*)d_in[2];
  const int*   dst     = (const int*)d_in[3];
  const float* rbf     = (const float*)d_in[4];
  const float* vecn    = (const float*)d_in[5];
  const float* emb_z   = (const float*)d_in[6];
  const float* ae_emb  = (const float*)d_in[7];
  const float* emb2_w  = (const float*)d_in[8];
  const float* emb2_b  = (const float*)d_in[9];
  const float* dp1_w   = (const float*)d_in[10];
  const float* dp1_b   = (const float*)d_in[11];
  const float* dp2_w   = (const float*)d_in[12];
  const float* dp2_b   = (const float*)d_in[13];
  const float* dp3_w   = (const float*)d_in[14];
  const float* dp3_b   = (const float*)d_in[15];
  const float* lt0     = (const float*)d_in[16];
  const float* lt1     = (const float*)d_in[17];
  const float* lt2     = (const float*)d_in[18];
  const float* ls1_w   = (const float*)d_in[19];
  const float* ls1_b   = (const float*)d_in[20];
  const float* ls2_w   = (const float*)d_in[21];
  const float* ls2_b   = (const float*)d_in[22];
  const float* ln_g    = (const float*)d_in[23];
  const float* ln_b    = (const float*)d_in[24];
  float* outp = (float*)d_out;

  // workspace layout (floats)
  float* Z     = (float*)d_ws;               // N*H
  float* nodeI = Z + (size_t)NN * HH;        // N*H
  float* nodeW = nodeI + (size_t)NN * HH;    // N*H*3
  float* nodeS = nodeW + (size_t)NN * HH * 3;// N*H*6
  float* lnout = nodeS + (size_t)NN * HH * 6;// N*H
  float* norm3 = lnout + (size_t)NN * HH;    // N*384

  // zero the 10 per-(n,h) accumulators (I + 3W + 6S), contiguous
  {
    int n4 = (NN * HH * 10) / 4;
    zero_kernel<<<(n4 + 255) / 256, 256, 0, stream>>>((float4*)nodeI, n4);
  }
  {
    int total = NN * HH;
    embed_kernel<<<(total + 255) / 256, 256, 0, stream>>>(x, ascal, emb_z, ae_emb, Z);
  }
  edge_kernel<<<EE / 16, 128, 0, stream>>>(Z, src, dst, rbf, vecn,
                                           emb2_w, emb2_b,
                                           dp1_w, dp1_b, dp2_w, dp2_b, dp3_w, dp3_b,
                                           nodeI, nodeW, nodeS);
  norm_kernel<<<NN, HH, 0, stream>>>(nodeI, nodeW, nodeS, ln_g, ln_b, lnout);
  mlp_kernel<<<NN / 16, 256, 0, stream>>>(lnout, ls1_w, ls1_b, ls2_w, ls2_b, norm3);
  out_kernel<<<NN / 16, 256, 0, stream>>>(nodeI, nodeW, nodeS, lt0, lt1, lt2, norm3, outp);
}